// Dynamics_59923383713896
// MI455X (gfx1250) — compile-verified
//
#include <hip/hip_runtime.h>
#include <hip/hip_bf16.h>
#include <math.h>

// ---------------------------------------------------------------------------
// EKF step for B=8192, XD=32, UD=16, AD=64, HD=128 on gfx1250 (MI455X).
// fp32 end-to-end using V_WMMA_F32_16X16X4_F32 (exact fp32 matrix cores).
// Pipeline: k_mlp(mean) -> k_prior -> k_mlp(prior_mean) -> k_post
// Intermediates staged in d_ws (~150MB, fp32).
// ---------------------------------------------------------------------------

typedef float v2f __attribute__((ext_vector_type(2)));
typedef float v8f __attribute__((ext_vector_type(8)));

#define BATCH 8192
#define XD 32
#define UD 16
#define AD 64
#define HD 128
#define MIN_VAR 0.01f
#define MAX_VAR 1.0f
#define EPSV 1e-6f

// workspace float offsets
#define OFF_A   0ull                               // 8192*1024  A coeffs (I+alpha applied)
#define OFF_B   (OFF_A  + 8192ull*1024ull)         // 8192*512   Bm coeffs
#define OFF_NX  (OFF_B  + 8192ull*512ull)          // 8192*32    nx variances
#define OFF_PM  (OFF_NX + 8192ull*32ull)           // 8192*32    prior mean
#define OFF_PC  (OFF_PM + 8192ull*32ull)           // 8192*1024  prior cov
#define OFF_C   (OFF_PC + 8192ull*1024ull)         // 8192*2048  C coeffs
#define OFF_NA  (OFF_C  + 8192ull*2048ull)         // 8192*64    na variances

// ---------------------------------------------------------------------------
// 16x16 fp32 WMMA tile: C += A(16xK) * B(Kx16), operands row-major in memory
// (LDS or global via generic pointers). AT/BT read the operand transposed.
// Fragment layout per CDNA5 ISA 7.12.2:
//   A 16x4: lanes0-15 M=lane, VGPR0/1 = K0,K1; lanes16-31 K2,K3
//   B 4x16 / C,D 16x16: row striped across lanes within a VGPR, halves split rows
// ---------------------------------------------------------------------------
template<bool AT, bool BT>
__device__ inline v8f wmma_tile_f32(const float* As, int lda,
                                    const float* Bs, int ldb,
                                    int K, v8f c) {
  const int lane = threadIdx.x & 31;
  const int koff = (lane >> 4) << 1;   // 0 for lanes 0-15, 2 for lanes 16-31
  const int mr   = lane & 15;          // A row / B col
  #pragma unroll 4
  for (int k0 = 0; k0 < K; k0 += 4) {
    const int ka = k0 + koff;
    v2f a, b;
    if (AT) { a.x = As[ka*lda + mr];  a.y = As[(ka+1)*lda + mr]; }
    else    { a.x = As[mr*lda + ka];  a.y = As[mr*lda + ka + 1]; }
    if (BT) { b.x = Bs[mr*ldb + ka];  b.y = Bs[mr*ldb + ka + 1]; }
    else    { b.x = Bs[ka*ldb + mr];  b.y = Bs[(ka+1)*ldb + mr]; }
    c = __builtin_amdgcn_wmma_f32_16x16x4_f32(false, a, false, b,
                                              (short)0, c, false, false);
  }
  return c;
}

// D-tile (16x16 f32, 8 VGPRs) scatter to row-major memory
__device__ inline void store_tile(float* dst, int ldd, v8f c) {
  const int lane = threadIdx.x & 31;
  const int col  = lane & 15;
  const int rb   = (lane >> 4) * 8;
  #pragma unroll
  for (int r = 0; r < 8; ++r) dst[(rb + r)*ldd + col] = c[r];
}

// ---------------------------------------------------------------------------
// MLP + coefficient heads. One wave handles 16 batch rows (batch = M dim).
// mode 0: heads = WA (I+alpha transform), WB (bias), Wnx (var transform)
// mode 1: heads = WC (bias), Wna (var transform)
// ---------------------------------------------------------------------------
#define MLP_WAVES 2

__global__ __launch_bounds__(32*MLP_WAVES)
void k_mlp(const float* __restrict__ x,
           const float* __restrict__ W1, const float* __restrict__ b1,
           const float* __restrict__ W2, const float* __restrict__ b2,
           const float* __restrict__ Wh0, const float* __restrict__ bh0,
           const float* __restrict__ Wh1, const float* __restrict__ bh1,
           const float* __restrict__ Wv,  const float* __restrict__ bv,
           const float* __restrict__ alpha_p,
           float* __restrict__ out0, float* __restrict__ out1,
           float* __restrict__ outv,
           int N0, int N1, int NV, int mode)
{
  __shared__ float Hs[MLP_WAVES][2][16*HD];           // 32KB static
  const int wave    = threadIdx.x >> 5;
  const int lane    = threadIdx.x & 31;
  const int rowbase = (blockIdx.x * MLP_WAVES + wave) * 16;
  float* H1 = Hs[wave][0];
  float* H2 = Hs[wave][1];
  const int col = lane & 15;
  const int rb  = (lane >> 4) * 8;

  // layer 1: relu(x @ W1 + b1), A-fragments straight from global x
  for (int n0 = 0; n0 < HD; n0 += 16) {
    v8f c = {};
    c = wmma_tile_f32<false,false>(x + (size_t)rowbase*XD, XD, W1 + n0, HD, XD, c);
    #pragma unroll
    for (int r = 0; r < 8; ++r) {
      float t = c[r] + b1[n0 + col];
      H1[(rb + r)*HD + n0 + col] = t > 0.f ? t : 0.f;
    }
  }
  // layer 2: relu(H1 @ W2 + b2)
  for (int n0 = 0; n0 < HD; n0 += 16) {
    v8f c = {};
    c = wmma_tile_f32<false,false>(H1, HD, W2 + n0, HD, HD, c);
    #pragma unroll
    for (int r = 0; r < 8; ++r) {
      float t = c[r] + b2[n0 + col];
      H2[(rb + r)*HD + n0 + col] = t > 0.f ? t : 0.f;
    }
  }

  const float alpha = alpha_p[0];

  // head 0: A-coeffs (mode0, I + alpha*(hWA+bA)) or C-coeffs (mode1)
  for (int n0 = 0; n0 < N0; n0 += 16) {
    v8f c = {};
    c = wmma_tile_f32<false,false>(H2, HD, Wh0 + n0, N0, HD, c);
    #pragma unroll
    for (int r = 0; r < 8; ++r) {
      const int f = n0 + col;
      float t = c[r] + bh0[f];
      float v = (mode == 0)
        ? (((f >> 5) == (f & 31)) ? 1.f : 0.f) + alpha * t
        : t;
      out0[(size_t)(rowbase + rb + r)*N0 + f] = v;
    }
  }
  // head 1: Bm coeffs (mode0 only)
  if (N1 > 0) {
    for (int n0 = 0; n0 < N1; n0 += 16) {
      v8f c = {};
      c = wmma_tile_f32<false,false>(H2, HD, Wh1 + n0, N1, HD, c);
      #pragma unroll
      for (int r = 0; r < 8; ++r) {
        const int f = n0 + col;
        out1[(size_t)(rowbase + rb + r)*N1 + f] = c[r] + bh1[f];
      }
    }
  }
  // variance head: MIN + (MAX-MIN)*sigmoid(hW + b)
  for (int n0 = 0; n0 < NV; n0 += 16) {
    v8f c = {};
    c = wmma_tile_f32<false,false>(H2, HD, Wv + n0, NV, HD, c);
    #pragma unroll
    for (int r = 0; r < 8; ++r) {
      const int f = n0 + col;
      float t = c[r] + bv[f];
      float s = 1.f / (1.f + __expf(-t));
      outv[(size_t)(rowbase + rb + r)*NV + f] = MIN_VAR + (MAX_VAR - MIN_VAR) * s;
    }
  }
}

// ---------------------------------------------------------------------------
// Prior update: one wave per batch element.
// prior_cov = psd(A cov A^T + Nx), prior_mean = A x + Bm u
// ---------------------------------------------------------------------------
__global__ __launch_bounds__(32)
void k_prior(const float* __restrict__ mean, const float* __restrict__ cov,
             const float* __restrict__ u, float* __restrict__ ws)
{
  __shared__ float Am[XD*XD], T1[XD*XD], T2[XD*XD];   // 12KB
  const int b    = blockIdx.x;
  const int lane = threadIdx.x;

  const float* cA = ws + OFF_A + (size_t)b*(XD*XD);
  for (int i = lane; i < XD*XD; i += 32) {
    Am[i] = cA[i];
    T1[i] = cov[(size_t)b*(XD*XD) + i];
  }
  __syncthreads();

  // T2 = A @ cov
  for (int mo = 0; mo < XD; mo += 16)
    for (int no = 0; no < XD; no += 16) {
      v8f c = {};
      c = wmma_tile_f32<false,false>(Am + mo*XD, XD, T1 + no, XD, XD, c);
      store_tile(T2 + mo*XD + no, XD, c);
    }
  __syncthreads();

  // T1 = T2 @ A^T
  for (int mo = 0; mo < XD; mo += 16)
    for (int no = 0; no < XD; no += 16) {
      v8f c = {};
      c = wmma_tile_f32<false,true>(T2 + mo*XD, XD, Am + no*XD, XD, XD, c);
      store_tile(T1 + mo*XD + no, XD, c);
    }
  __syncthreads();

  // prior mean (lane i = row i)
  const float* cB = ws + OFF_B + (size_t)b*(XD*UD);
  const float nxv = ws[OFF_NX + (size_t)b*XD + lane];
  {
    float pm = 0.f;
    for (int j = 0; j < XD; ++j) pm += Am[lane*XD + j] * mean[(size_t)b*XD + j];
    for (int j = 0; j < UD; ++j) pm += cB[lane*UD + j] * u[(size_t)b*UD + j];
    ws[OFF_PM + (size_t)b*XD + lane] = pm;
  }
  // prior cov: symmetrize + Nx + eps
  float* pc = ws + OFF_PC + (size_t)b*(XD*XD);
  for (int j = 0; j < XD; ++j) {
    float v = 0.5f*(T1[lane*XD + j] + T1[j*XD + lane]);
    if (lane == j) v += nxv + EPSV;
    pc[lane*XD + j] = v;
  }
}

// ---------------------------------------------------------------------------
// Measurement update: one wave per batch element.
// S = C P C^T + Na (SPD) ; Cholesky ; Y = S^-1 (C P) ; G = Y^T
// post_mean = pm + G (a - C pm) ; post_cov = psd(P - Y^T (C P))
// ---------------------------------------------------------------------------
__global__ __launch_bounds__(32)
void k_post(const float* __restrict__ a, float* __restrict__ ws,
            float* __restrict__ out)
{
  __shared__ float Cm[AD*XD];      // 64x32
  __shared__ float P [XD*XD];
  __shared__ float CP[AD*XD];
  __shared__ float S [AD*AD];      // also holds L, then reused for D=Y^T CP
  __shared__ float Y [AD*XD];
  __shared__ float vecs[XD + AD];  // prior_mean | innovation
  const int b    = blockIdx.x;
  const int lane = threadIdx.x;

  const float* cC = ws + OFF_C  + (size_t)b*(AD*XD);
  const float* pc = ws + OFF_PC + (size_t)b*(XD*XD);
  for (int i = lane; i < AD*XD; i += 32) Cm[i] = cC[i];
  for (int i = lane; i < XD*XD; i += 32) P[i]  = pc[i];
  __syncthreads();

  // CP = C @ P
  for (int mo = 0; mo < AD; mo += 16)
    for (int no = 0; no < XD; no += 16) {
      v8f c = {};
      c = wmma_tile_f32<false,false>(Cm + mo*XD, XD, P + no, XD, XD, c);
      store_tile(CP + mo*XD + no, XD, c);
    }
  __syncthreads();

  // S = CP @ C^T
  for (int mo = 0; mo < AD; mo += 16)
    for (int no = 0; no < AD; no += 16) {
      v8f c = {};
      c = wmma_tile_f32<false,true>(CP + mo*XD, XD, Cm + no*XD, XD, XD, c);
      store_tile(S + mo*AD + no, AD, c);
    }
  __syncthreads();
  for (int i = lane; i < AD; i += 32)
    S[i*AD + i] += ws[OFF_NA + (size_t)b*AD + i];
  __syncthreads();

  // In-LDS Cholesky (lower triangle of S becomes L)
  for (int k = 0; k < AD; ++k) {
    const float lkk = sqrtf(S[k*AD + k]);
    const float inv = 1.f / lkk;
    if (lane == 0) S[k*AD + k] = lkk;
    __syncthreads();
    for (int i = k + 1 + lane; i < AD; i += 32) S[i*AD + k] *= inv;
    __syncthreads();
    for (int i = k + 1 + lane; i < AD; i += 32) {
      const float lik = S[i*AD + k];
      for (int j = k + 1; j <= i; ++j) S[i*AD + j] -= lik * S[j*AD + k];
    }
    __syncthreads();
  }

  // Y = S^-1 CP : forward/backward triangular solves, lane owns one column
  for (int i = lane; i < AD*XD; i += 32) Y[i] = CP[i];
  __syncthreads();
  {
    const int cc = lane;
    for (int k = 0; k < AD; ++k) {
      const float yk = Y[k*XD + cc] / S[k*AD + k];
      Y[k*XD + cc] = yk;
      for (int i = k + 1; i < AD; ++i) Y[i*XD + cc] -= S[i*AD + k] * yk;
    }
    for (int k = AD - 1; k >= 0; --k) {
      const float yk = Y[k*XD + cc] / S[k*AD + k];
      Y[k*XD + cc] = yk;
      for (int i = 0; i < k; ++i) Y[i*XD + cc] -= S[k*AD + i] * yk;
    }
  }
  __syncthreads();

  // innovation + post_mean
  float* pmS   = vecs;
  float* innov = vecs + XD;
  pmS[lane] = ws[OFF_PM + (size_t)b*XD + lane];
  __syncthreads();
  for (int j = lane; j < AD; j += 32) {
    float acc = a[(size_t)b*AD + j];
    for (int k = 0; k < XD; ++k) acc -= Cm[j*XD + k] * pmS[k];
    innov[j] = acc;
  }
  __syncthreads();
  {
    float pm = pmS[lane];
    for (int j = 0; j < AD; ++j) pm += Y[j*XD + lane] * innov[j];   // G = Y^T
    out[(size_t)b*XD + lane] = pm;
  }
  __syncthreads();

  // D = Y^T @ CP (32x32), reuse S as scratch
  for (int mo = 0; mo < XD; mo += 16)
    for (int no = 0; no < XD; no += 16) {
      v8f c = {};
      c = wmma_tile_f32<true,false>(Y + mo, XD, CP + no, XD, AD, c);
      store_tile(S + mo*XD + no, XD, c);
    }
  __syncthreads();

  // post_cov = psd(P - D)
  float* outc = out + (size_t)BATCH*XD + (size_t)b*(XD*XD);
  for (int j = 0; j < XD; ++j) {
    const float v = P[lane*XD + j] - S[lane*XD + j];
    const float w = P[j*XD + lane] - S[j*XD + lane];
    float r = 0.5f*(v + w);
    if (lane == j) r += EPSV;
    outc[lane*XD + j] = r;
  }
}

// ---------------------------------------------------------------------------
extern "C" void kernel_launch(void* const* d_in, const int* in_sizes, int n_in,
                              void* d_out, int out_size, void* d_ws, size_t ws_size,
                              hipStream_t stream)
{
  const float* mean  = (const float*)d_in[0];
  const float* cov   = (const float*)d_in[1];
  const float* u     = (const float*)d_in[2];
  const float* a     = (const float*)d_in[3];
  const float* W1    = (const float*)d_in[4];
  const float* b1    = (const float*)d_in[5];
  const float* W2    = (const float*)d_in[6];
  const float* b2    = (const float*)d_in[7];
  const float* WA    = (const float*)d_in[8];
  const float* bA    = (const float*)d_in[9];
  const float* WB    = (const float*)d_in[10];
  const float* bB    = (const float*)d_in[11];
  const float* WC    = (const float*)d_in[12];
  const float* bC    = (const float*)d_in[13];
  const float* Wnx   = (const float*)d_in[14];
  const float* bnx   = (const float*)d_in[15];
  const float* Wna   = (const float*)d_in[16];
  const float* bna   = (const float*)d_in[17];
  const float* alpha = (const float*)d_in[18];
  float* ws  = (float*)d_ws;
  float* out = (float*)d_out;

  const dim3 mlpBlk(32*MLP_WAVES);
  const dim3 mlpGrd(BATCH / (16*MLP_WAVES));

  // pass 1: dynamics at current mean -> A, Bm, nx
  k_mlp<<<mlpGrd, mlpBlk, 0, stream>>>(mean, W1,b1, W2,b2, WA,bA, WB,bB,
                                       Wnx,bnx, alpha,
                                       ws+OFF_A, ws+OFF_B, ws+OFF_NX,
                                       XD*XD, XD*UD, XD, 0);
  // time update
  k_prior<<<dim3(BATCH), dim3(32), 0, stream>>>(mean, cov, u, ws);
  // pass 2: dynamics at prior mean -> C, na
  k_mlp<<<mlpGrd, mlpBlk, 0, stream>>>(ws+OFF_PM, W1,b1, W2,b2, WC,bC,
                                       nullptr,nullptr, Wna,bna, alpha,
                                       ws+OFF_C, nullptr, ws+OFF_NA,
                                       AD*XD, 0, AD, 1);
  // measurement update
  k_post<<<dim3(BATCH), dim3(32), 0, stream>>>(a, ws, out);
}